// RAYBENDS_40192303956377
// MI455X (gfx1250) — compile-verified
//
#include <hip/hip_runtime.h>

typedef float v2f __attribute__((ext_vector_type(2)));
typedef float v8f __attribute__((ext_vector_type(8)));

#define NBLOCKS 512
#define NTHREADS 256
#define WAVES_PER_BLOCK (NTHREADS / 32)

// Cubic B-spline basis rows: row 0..3 correspond to knot offsets -2..+1
__device__ __forceinline__ float qrow(int row, float t) {
    float t2 = t * t;
    float t3 = t2 * t;
    switch (row) {
        case 0:  return 1.0f/6.0f - 0.5f*t + 0.5f*t2 - t3*(1.0f/6.0f);
        case 1:  return 2.0f/3.0f - t2 + 0.5f*t3;
        case 2:  return 1.0f/6.0f + 0.5f*t + 0.5f*t2 - 0.5f*t3;
        default: return t3*(1.0f/6.0f);
    }
}

// Per-segment integrand: 0.5*(1/c0 + 1/c1) * sqrt(dx^2 + dz^2)
// with one hardware rcp and one hardware sqrt (both ~1 ulp).
__device__ __forceinline__ float seg_term(float zv, float dzv, float dxv) {
    float c0   = __builtin_fmaf(0.01f, zv,  1.0f);
    float c1   = __builtin_fmaf(0.01f, dzv, c0);
    float havg = __builtin_fmaf(0.005f, dzv, c0);       // 0.5*(c0+c1)
    float prod = c0 * c1;
    float sl   = havg * __builtin_amdgcn_rcpf(prod);    // 0.5*(c0+c1)/(c0*c1)
    float d2   = __builtin_fmaf(dzv, dzv, dxv * dxv);
    return sl * __builtin_amdgcn_sqrtf(d2);
}

__global__ __launch_bounds__(NTHREADS)
void raybend_pass1(const float* __restrict__ y, int nknots,
                   float* __restrict__ partials) {
    const int lane  = threadIdx.x & 31;
    const int wave  = threadIdx.x >> 5;
    const int gwave = blockIdx.x * WAVES_PER_BLOCK + wave;
    const int nwaves = gridDim.x * WAVES_PER_BLOCK;
    const int nrows  = nknots + 2;
    const int nfull  = nrows >> 4;       // full 16-row tiles (no row masking)
    const int ntail  = nrows & 15;       // rows in the final partial tile

    const int Mn   = lane & 15;   // M for A-matrix, N for B/D matrices
    const int half = lane >> 4;
    const int kb   = half * 2;    // K base held by this lane (A and B layout)

    // B-matrix operands (4x16 f32, 2 VGPRs/lane):
    //  Bq : columns 0..10 = q(t_j),               rest 0  (spline values)
    //  Bd : columns 0..9  = q(t_{j+1}) - q(t_j),  rest 0  (segment deltas)
    // Columns >= 10 of Bd are zero => dist = 0 => those lanes contribute
    // exactly 0; no column masking needed anywhere.
    v2f Bq = {0.0f, 0.0f};
    v2f Bd = {0.0f, 0.0f};
    if (Mn <= 10) {
        float t = (float)Mn * 0.1f;
        Bq.x = qrow(kb,     t);
        Bq.y = qrow(kb + 1, t);
        if (Mn <= 9) {
            float tn = (float)(Mn + 1) * 0.1f;
            Bd.x = qrow(kb,     tn) - Bq.x;
            Bd.y = qrow(kb + 1, tn) - Bq.y;
        }
    }

    const v2f* __restrict__ y2 = (const v2f*)y;  // y2[k] = (x[k], z[k])

    float acc = 0.0f;
    for (int tile = gwave; tile < nfull; tile += nwaves) {
        const int r0 = tile << 4;

        // A-matrix operands: A[M][K] = x[clamp(r0 + M - 2 + K)] (and z for Z)
        int k0 = r0 + Mn - 2 + kb;
        int k1 = k0 + 1;
        k0 = min(max(k0, 0), nknots);
        k1 = min(max(k1, 0), nknots);
        v2f p0 = y2[k0];
        v2f p1 = y2[k1];
        v2f aX = {p0.x, p1.x};
        v2f aZ = {p0.y, p1.y};

        v8f zs = {};   // zs[j]
        v8f dz = {};   // zs[j+1]-zs[j]
        v8f dx = {};   // xs[j+1]-xs[j]
        zs = __builtin_amdgcn_wmma_f32_16x16x4_f32(
                 false, aZ, false, Bq, (short)0, zs, false, false);
        dz = __builtin_amdgcn_wmma_f32_16x16x4_f32(
                 false, aZ, false, Bd, (short)0, dz, false, false);
        dx = __builtin_amdgcn_wmma_f32_16x16x4_f32(
                 false, aX, false, Bd, (short)0, dx, false, false);

        // D layout: element r = row (half*8 + r), column N = lane&15.
        // Full tile: every row valid, zero-basis columns self-cancel.
#pragma unroll
        for (int r = 0; r < 8; ++r)
            acc += seg_term(zs[r], dz[r], dx[r]);
    }

    // Final partial tile (rows nfull*16 .. nrows-1), done once by wave 0.
    // Branch is wave-uniform, so EXEC is all-ones at the WMMAs.
    if (gwave == 0 && ntail != 0) {
        const int r0 = nfull << 4;
        int k0 = r0 + Mn - 2 + kb;
        int k1 = k0 + 1;
        k0 = min(max(k0, 0), nknots);
        k1 = min(max(k1, 0), nknots);
        v2f p0 = y2[k0];
        v2f p1 = y2[k1];
        v2f aX = {p0.x, p1.x};
        v2f aZ = {p0.y, p1.y};

        v8f zs = {};
        v8f dz = {};
        v8f dx = {};
        zs = __builtin_amdgcn_wmma_f32_16x16x4_f32(
                 false, aZ, false, Bq, (short)0, zs, false, false);
        dz = __builtin_amdgcn_wmma_f32_16x16x4_f32(
                 false, aZ, false, Bd, (short)0, dz, false, false);
        dx = __builtin_amdgcn_wmma_f32_16x16x4_f32(
                 false, aX, false, Bd, (short)0, dx, false, false);

#pragma unroll
        for (int r = 0; r < 8; ++r) {
            int rowIdx = r0 + half * 8 + r;
            float term = seg_term(zs[r], dz[r], dx[r]);
            acc += (rowIdx < nrows) ? term : 0.0f;
        }
    }

    // Deterministic wave reduction (fixed tree), then fixed-order block sum.
#pragma unroll
    for (int off = 16; off >= 1; off >>= 1)
        acc += __shfl_down(acc, off, 32);

    __shared__ float wsum[WAVES_PER_BLOCK];
    if (lane == 0) wsum[wave] = acc;
    __syncthreads();
    if (threadIdx.x == 0) {
        float s = 0.0f;
        for (int w = 0; w < WAVES_PER_BLOCK; ++w) s += wsum[w];
        partials[blockIdx.x] = s;
    }
}

__global__ __launch_bounds__(NBLOCKS)
void raybend_pass2(const float* __restrict__ partials, float* __restrict__ out) {
    __shared__ float sm[NBLOCKS];
    int t = threadIdx.x;
    sm[t] = partials[t];
    __syncthreads();
    for (int s = NBLOCKS / 2; s > 0; s >>= 1) {
        if (t < s) sm[t] += sm[t + s];
        __syncthreads();
    }
    if (t == 0) out[0] = sm[0];
}

extern "C" void kernel_launch(void* const* d_in, const int* in_sizes, int n_in,
                              void* d_out, int out_size, void* d_ws, size_t ws_size,
                              hipStream_t stream) {
    const float* y = (const float*)d_in[0];
    // y has 2*(nknots+1) elements -> recover nknots on the host from in_sizes.
    int nknots = in_sizes[0] / 2 - 1;
    float* partials = (float*)d_ws;  // needs NBLOCKS * 4 bytes of scratch

    raybend_pass1<<<NBLOCKS, NTHREADS, 0, stream>>>(y, nknots, partials);
    raybend_pass2<<<1, NBLOCKS, 0, stream>>>(partials, (float*)d_out);
}